// LenPredictorAttention_20401094656286
// MI455X (gfx1250) — compile-verified
//
#include <hip/hip_runtime.h>
#include <hip/hip_bf16.h>

// Problem constants (fixed by the reference).
#define NBATCH   16
#define ENC_T    512
#define ENC_DIM  256
#define DEC_T    4096

#define THREADS        512           // 16 wave32s
#define NWAVES         (THREADS / 32)
#define ROWS_PER_BLOCK 64            // decoder rows per block (4 per wave)

// Native vector types (HIP's float4 class is rejected by clang builtins).
typedef int   v4i __attribute__((vector_size(16)));   // matches builtin param
typedef float v4f __attribute__((ext_vector_type(4)));

// gfx1250 async global->LDS path (ASYNCcnt-tracked), with safe fallback.
#if defined(__gfx1250__) && __has_builtin(__builtin_amdgcn_global_load_async_to_lds_b128)
#define USE_ASYNC_LDS 1
#else
#define USE_ASYNC_LDS 0
#endif

__global__ __launch_bounds__(THREADS) void lenreg_gather_kernel(
    const float* __restrict__ enc,   // [B, ENC_T, ENC_DIM]
    const float* __restrict__ lens,  // [B, ENC_T] (integer-valued fp32 durations)
    float* __restrict__ out)         // [B, DEC_T, ENC_DIM]
{
    __shared__ float s_len[ENC_T];   // raw durations
    __shared__ int   s_cum[ENC_T];   // inclusive cumsum (end positions)

    const int tid = threadIdx.x;
    const int b   = blockIdx.x / (DEC_T / ROWS_PER_BLOCK);
    const int t0  = (blockIdx.x % (DEC_T / ROWS_PER_BLOCK)) * ROWS_PER_BLOCK;

    // ---- Stage encoder_lengths[b,:] (2 KB) into LDS ----------------------
#if USE_ASYNC_LDS
    if (tid < ENC_T / 4) {
        const float* gsrc = lens + (size_t)b * ENC_T + tid * 4;
        // Two-step cast: addrspace(3) first, then pointee type.
        __attribute__((address_space(3))) float* lds_f =
            (__attribute__((address_space(3))) float*)&s_len[tid * 4];
        __attribute__((address_space(3))) v4i* lds_v =
            (__attribute__((address_space(3))) v4i*)lds_f;
        __builtin_amdgcn_global_load_async_to_lds_b128(
            (v4i*)gsrc, lds_v, /*imm offset*/ 0, /*cpol*/ 0);
    }
#if __has_builtin(__builtin_amdgcn_s_wait_asynccnt)
    __builtin_amdgcn_s_wait_asynccnt(0);
#else
    asm volatile("s_wait_asynccnt 0" ::: "memory");
#endif
#else
    s_len[tid] = lens[(size_t)b * ENC_T + tid];
#endif
    __syncthreads();

    // ---- Inclusive scan (Hillis-Steele, 512 elems, exact in int) ---------
    s_cum[tid] = (int)s_len[tid];
    __syncthreads();
#pragma unroll
    for (int off = 1; off < ENC_T; off <<= 1) {
        int x = s_cum[tid];
        if (tid >= off) x += s_cum[tid - off];
        __syncthreads();
        s_cum[tid] = x;
        __syncthreads();
    }
    const int total = s_cum[ENC_T - 1];

    // ---- Gather: one output row (1 KB) per wave per iteration ------------
    const int wave = tid >> 5;
    const int lane = tid & 31;

    for (int r = wave; r < ROWS_PER_BLOCK; r += NWAVES) {
        const int t = t0 + r;

        v4f v0 = {0.f, 0.f, 0.f, 0.f};
        v4f v1 = {0.f, 0.f, 0.f, 0.f};

        if (t < total) {
            // first i with cum[i] > t  (then cum[i-1] = start_i <= t < cum[i])
            int lo = 0, hi = ENC_T;
            while (lo < hi) {
                int mid = (lo + hi) >> 1;
                if (s_cum[mid] <= t) lo = mid + 1; else hi = mid;
            }
            const v4f* src =
                (const v4f*)(enc + ((size_t)(b * ENC_T + lo)) * ENC_DIM);
            v0 = src[lane];        // global_load_b128
            v1 = src[lane + 32];   // global_load_b128
        }

        // Streaming (non-temporal) 16B stores; output is write-once.
        v4f* dst = (v4f*)(out + ((size_t)(b * DEC_T + t)) * ENC_DIM);
        __builtin_nontemporal_store(v0, dst + lane);
        __builtin_nontemporal_store(v1, dst + lane + 32);
    }
}

extern "C" void kernel_launch(void* const* d_in, const int* in_sizes, int n_in,
                              void* d_out, int out_size, void* d_ws, size_t ws_size,
                              hipStream_t stream) {
    const float* enc  = (const float*)d_in[0];  // encoder_outputs fp32 [16,512,256]
    const float* lens = (const float*)d_in[1];  // encoder_lengths fp32 [16,512]
    // d_in[2] (output_lengths) and d_in[3] (dec_T) are fixed by the problem; unused.
    float* out = (float*)d_out;                 // [16,4096,256] fp32

    const dim3 grid(NBATCH * (DEC_T / ROWS_PER_BLOCK));  // 1024 blocks
    lenreg_gather_kernel<<<grid, THREADS, 0, stream>>>(enc, lens, out);
}